// GCNGraphEncoder_2327872274735
// MI455X (gfx1250) — compile-verified
//
#include <hip/hip_runtime.h>

#define NN   100000
#define EE   1600000
#define FF   128
#define CC   64
#define NHID 256
#define NOUTD 128
#define LL   4
#define GG   512

typedef __attribute__((ext_vector_type(2))) float v2f;
typedef __attribute__((ext_vector_type(8))) float v8f;

__device__ __forceinline__ void atomicAddF(float* p, float v) {
    // relaxed, agent-scope, non-returning -> global_atomic_add_f32 (STOREcnt path)
    __hip_atomic_fetch_add(p, v, __ATOMIC_RELAXED, __HIP_MEMORY_SCOPE_AGENT);
}

// ---- degree count (int atomics into dinv buffer reinterpreted as u32) ----
__global__ __launch_bounds__(256)
void k_degree(const int* __restrict__ dst, unsigned* __restrict__ cnt) {
    int e = blockIdx.x * 256 + threadIdx.x;
    if (e < EE) atomicAdd(&cnt[dst[e]], 1u);
}

// in-place: u32 count -> rsqrt(count + 1)
__global__ __launch_bounds__(256)
void k_dinv(float* __restrict__ buf) {
    int i = blockIdx.x * 256 + threadIdx.x;
    if (i < NN) {
        unsigned c = ((const unsigned*)buf)[i];
        buf[i] = rsqrtf((float)c + 1.0f);
    }
}

// ---- f32 WMMA GEMM: Out[M,Cn] = A[M,K] @ B[K,Cn] (+ optional bias / relu) ----
// Block = 128 threads (4 waves) -> 32 rows x 64 cols of output.
// Each wave: 32x16 patch = two independent 16x16 accumulators (breaks the
// WMMA RAW chain, reuses B across both tiles). B's 64-col slice is staged
// once per block into LDS, transposed with +2 padding so each lane's K-pair
// is one aligned ds_load_b64.
// Requires: M % 32 == 0, Cn % 64 == 0, K % 4 == 0 (all true here).
__global__ __launch_bounds__(128)
void k_gemm_f32(const float* __restrict__ A, const float* __restrict__ B,
                const float* __restrict__ bias, float* __restrict__ Out,
                int M, int K, int Cn, int relu)
{
    extern __shared__ float Bl[];              // [64][K+2] = B slice, transposed
    const int tid  = threadIdx.x;
    const int lane = tid & 31;
    const int wave = tid >> 5;
    const int nblk = Cn >> 6;
    const int bm   = blockIdx.x / nblk;
    const int bn   = blockIdx.x % nblk;
    const int ldb  = K + 2;

    // cooperative stage: B[k][bn*64 + n] -> Bl[n][k]
    for (int idx = tid; idx < (K << 6); idx += 128) {
        int k = idx >> 6, n = idx & 63;
        Bl[n * ldb + k] = B[(long)k * Cn + (bn << 6) + n];
    }
    __syncthreads();

    const int lo   = lane & 15, hi = lane >> 4;
    const int koff = hi << 1;                  // ISA: K split across lane halves
    const long row0 = (long)(bm << 5) + lo;    // A rows row0 and row0+16
    const int  nloc = (wave << 4) + lo;        // local col 0..63
    const int  ncol = (bn << 6) + nloc;        // global col

    const float* A0 = A + row0 * (long)K + koff;
    const float* A1 = A0 + (long)16 * K;
    const float* Bp = Bl + nloc * ldb + koff;

    v8f acc0 = {}, acc1 = {};
    for (int k = 0; k < K; k += 4) {
        v2f a0, a1, b;
        a0.x = A0[k]; a0.y = A0[k + 1];        // global_load_b64
        a1.x = A1[k]; a1.y = A1[k + 1];        // global_load_b64
        b.x  = Bp[k]; b.y  = Bp[k + 1];        // ds_load_b64
        acc0 = __builtin_amdgcn_wmma_f32_16x16x4_f32(
                   false, a0, false, b, (short)0, acc0, false, false);
        acc1 = __builtin_amdgcn_wmma_f32_16x16x4_f32(
                   false, a1, false, b, (short)0, acc1, false, false);
    }

    const float bv = bias ? bias[ncol] : 0.0f;
#pragma unroll
    for (int r = 0; r < 8; ++r) {              // D: VGPR r -> row r + 8*hi
        long rr = (long)(bm << 5) + r + (hi << 3);
        float v0 = acc0[r] + bv;
        float v1 = acc1[r] + bv;
        if (relu) { v0 = fmaxf(v0, 0.0f); v1 = fmaxf(v1, 0.0f); }
        Out[rr * (long)Cn + ncol]        = v0;
        Out[(rr + 16) * (long)Cn + ncol] = v1;
    }
}

// ---- edge aggregation: agg[dst] += tmp[src] * dinv[src]*dinv[dst] ----
// 16 lanes per edge, float4 gather, f32 atomic scatter (L2-resident)
__global__ __launch_bounds__(256)
void k_edge_agg(const int* __restrict__ src, const int* __restrict__ dst,
                const float* __restrict__ dinv, const float* __restrict__ tmp,
                float* __restrict__ agg)
{
    long t = (long)blockIdx.x * 256 + threadIdx.x;
    int e = (int)(t >> 4);
    int q = (int)(t & 15);
    if (e >= EE) return;
    int s = src[e], d = dst[e];
    float coef = dinv[s] * dinv[d];
    float4 v = ((const float4*)(tmp + (long)s * CC))[q];
    float* p = agg + (long)d * CC + (q << 2);
    atomicAddF(p + 0, v.x * coef);
    atomicAddF(p + 1, v.y * coef);
    atomicAddF(p + 2, v.z * coef);
    atomicAddF(p + 3, v.w * coef);
}

// ---- h = relu(agg + tmp*dinv^2 + bias) ----
__global__ __launch_bounds__(256)
void k_finalize(const float* __restrict__ tmp, const float* __restrict__ agg,
                const float* __restrict__ dinv, const float* __restrict__ bias,
                float* __restrict__ hout)
{
    int i = blockIdx.x * 256 + threadIdx.x;
    if (i < NN * CC) {
        int node = i >> 6, c = i & 63;
        float di = dinv[node];
        float v = agg[i] + tmp[i] * di * di + bias[c];
        hout[i] = fmaxf(v, 0.0f);
    }
}

// ---- global mean pool (accumulate) ----
__global__ __launch_bounds__(256)
void k_pool(const int* __restrict__ batch, const float* __restrict__ h,
            float* __restrict__ pooled, unsigned* __restrict__ counts)
{
    long t = (long)blockIdx.x * 256 + threadIdx.x;
    int node = (int)(t >> 4);
    int q    = (int)(t & 15);
    if (node >= NN) return;
    int g = batch[node];
    if (q == 0) atomicAdd(&counts[g], 1u);
    float4 v = ((const float4*)(h + (long)node * CC))[q];
    float* p = pooled + (long)g * CC + (q << 2);
    atomicAddF(p + 0, v.x);
    atomicAddF(p + 1, v.y);
    atomicAddF(p + 2, v.z);
    atomicAddF(p + 3, v.w);
}

__global__ __launch_bounds__(256)
void k_divide(float* __restrict__ pooled, const unsigned* __restrict__ counts) {
    int i = blockIdx.x * 256 + threadIdx.x;
    if (i < GG * CC)
        pooled[i] /= fmaxf((float)counts[i >> 6], 1.0f);
}

extern "C" void kernel_launch(void* const* d_in, const int* in_sizes, int n_in,
                              void* d_out, int out_size, void* d_ws, size_t ws_size,
                              hipStream_t stream)
{
    (void)in_sizes; (void)n_in; (void)out_size; (void)ws_size;
    const float* x   = (const float*)d_in[0];
    const int*   ei  = (const int*)d_in[1];      // [2,E] flat: src then dst
    const int*   bat = (const int*)d_in[2];
    const float* W0  = (const float*)d_in[3];
    const float* b0  = (const float*)d_in[4];
    const float* Wh  = (const float*)d_in[5];
    const float* bh  = (const float*)d_in[6];
    const float* W1  = (const float*)d_in[7];
    const float* b1  = (const float*)d_in[8];
    const float* W2  = (const float*)d_in[9];
    const float* b2  = (const float*)d_in[10];
    float*       out = (float*)d_out;

    const int* src = ei;
    const int* dst = ei + EE;

    // workspace layout (floats): dinv | h | tmp | agg | pooled | counts | hid
    float*    ws     = (float*)d_ws;
    float*    dinv   = ws;
    float*    bufA   = ws + NN;                        // h  (N x C)
    float*    bufB   = bufA + (long)NN * CC;           // tmp = h @ W (N x C)
    float*    bufC   = bufB + (long)NN * CC;           // agg (N x C)
    float*    pooled = bufC + (long)NN * CC;           // G x C
    unsigned* counts = (unsigned*)(pooled + GG * CC);  // G
    float*    hid    = (float*)(counts + GG);          // G x NH

    // degree + dinv
    hipMemsetAsync(dinv, 0, NN * sizeof(float), stream);
    k_degree<<<(EE + 255) / 256, 256, 0, stream>>>(dst, (unsigned*)dinv);
    k_dinv<<<(NN + 255) / 256, 256, 0, stream>>>(dinv);

    const int aggGrid = (int)(((long)EE * 16) / 256);      // 100000
    const int finGrid = (NN * CC + 255) / 256;

    // GEMM launch helper values: grid = (M/32)*(Cn/64), lds = 64*(K+2)*4
    const int nodeGrid   = (NN / 32) * (CC / 64);          // 3125
    const size_t lds64   = 64 * (64  + 2) * sizeof(float); // K=64
    const size_t lds128  = 64 * (128 + 2) * sizeof(float); // K=128
    const size_t lds256  = 64 * (256 + 2) * sizeof(float); // K=256

    // layer 0: 128 -> 64
    k_gemm_f32<<<nodeGrid, 128, lds128, stream>>>(x, W0, nullptr, bufB, NN, FF, CC, 0);
    hipMemsetAsync(bufC, 0, (size_t)NN * CC * sizeof(float), stream);
    k_edge_agg<<<aggGrid, 256, 0, stream>>>(src, dst, dinv, bufB, bufC);
    k_finalize<<<finGrid, 256, 0, stream>>>(bufB, bufC, dinv, b0, bufA);

    // hidden layers: 64 -> 64
    for (int l = 0; l < LL; ++l) {
        k_gemm_f32<<<nodeGrid, 128, lds64, stream>>>(bufA, Wh + (long)l * CC * CC,
                                                     nullptr, bufB, NN, CC, CC, 0);
        hipMemsetAsync(bufC, 0, (size_t)NN * CC * sizeof(float), stream);
        k_edge_agg<<<aggGrid, 256, 0, stream>>>(src, dst, dinv, bufB, bufC);
        k_finalize<<<finGrid, 256, 0, stream>>>(bufB, bufC, dinv, bh + l * CC, bufA);
    }

    // mean pool over graphs
    hipMemsetAsync(pooled, 0, (GG * CC + GG) * sizeof(float), stream);
    k_pool<<<(NN * 16 + 255) / 256, 256, 0, stream>>>(bat, bufA, pooled, counts);
    k_divide<<<(GG * CC + 255) / 256, 256, 0, stream>>>(pooled, counts);

    // MLP head: relu(pooled @ W1 + b1) @ W2 + b2
    k_gemm_f32<<<(GG / 32) * (NHID / 64), 128, lds64, stream>>>(
        pooled, W1, b1, hid, GG, CC, NHID, 1);
    k_gemm_f32<<<(GG / 32) * (NOUTD / 64), 128, lds256, stream>>>(
        hid, W2, b2, out, GG, NHID, NOUTD, 0);
}